// GDN_16965120819904
// MI455X (gfx1250) — compile-verified
//
#include <hip/hip_runtime.h>
#include <hip/hip_bf16.h>
#include <math.h>

// GDN forward for MI455X (gfx1250, wave32).
// Shapes: B=128, N=256, W=64, D=128, K=20.
// f32 GEMMs use V_WMMA_F32_16X16X4_F32 with register blocking (cov/cos: 16x64
// tiles, 4 accumulators per wave) and LDS-staged B (lin: 16x128 strip per wave,
// 8 accumulators, lin_W resident in LDS).

typedef __attribute__((ext_vector_type(2))) float v2f;
typedef __attribute__((ext_vector_type(8))) float v8f;

#define NB 128   // batch
#define NN 256   // nodes
#define NW 64    // window
#define ND 128   // embedding dim
#define NK 20    // top-k
#define NC 8192  // B*W (row length of xr)

// ---------------------------------------------------------------------------
// 1) per-node mean of xr row (8192 elems) and L2 norm of emb row (128 elems)
__global__ void gdn_stats(const float* __restrict__ data, const float* __restrict__ emb,
                          float* __restrict__ mean, float* __restrict__ nrm) {
  __shared__ float red[256];
  const int i = blockIdx.x;          // node
  const int t = threadIdx.x;
  float s = 0.f;
  for (int c = t; c < NC; c += 256) {
    int b = c >> 6, w = c & 63;      // xr[i][c] = data[b][i][w]
    s += data[(b * NN + i) * NW + w];
  }
  red[t] = s; __syncthreads();
  for (int off = 128; off > 0; off >>= 1) { if (t < off) red[t] += red[t + off]; __syncthreads(); }
  if (t == 0) mean[i] = red[0] * (1.0f / (float)NC);
  __syncthreads();
  float q = 0.f;
  if (t < ND) { float v = emb[i * ND + t]; q = v * v; }
  red[t] = q; __syncthreads();
  for (int off = 128; off > 0; off >>= 1) { if (t < off) red[t] += red[t + off]; __syncthreads(); }
  if (t == 0) nrm[i] = sqrtf(red[0]);
}

// ---------------------------------------------------------------------------
// 2) one wave per 16x64 strip of the [256x256] similarity matrices.
//    Register blocking: one A fragment feeds 4 WMMAs (4 C accumulators).
//    accE = emb @ emb^T (K=128), accC = xc @ xc^T (K=8192, centered on the fly)
__global__ void gdn_simgemm(const float* __restrict__ data, const float* __restrict__ emb,
                            const float* __restrict__ mean,
                            float* __restrict__ cosd, float* __restrict__ cov) {
  const int ti  = blockIdx.x >> 2;           // 16 row tiles
  const int tjg = blockIdx.x & 3;            // 4 col groups of 64
  const int lane = threadIdx.x & 31;
  const int i0 = ti * 16, j0 = tjg * 64;
  const int m = lane & 15;                   // A row / B col within tile
  const int koff = (lane >> 4) << 1;         // K sub-offset {0,2}
  const int rA = i0 + m;
  const float meanA = mean[rA];
  int   nB[4];
  float meanB[4];
#pragma unroll
  for (int q = 0; q < 4; ++q) { nB[q] = j0 + q * 16 + m; meanB[q] = mean[nB[q]]; }

  const int rowbase = i0 + ((lane >> 4) << 3);   // VGPR g -> row g + 8*(lane>>4)
  const int col0 = (lane & 15);

  // ---- cos: emb @ emb^T, K = 128 ----
  {
    v8f acc[4] = {};
#pragma unroll 1
    for (int kk = 0; kk < ND; kk += 4) {
      v2f a;
      a.x = emb[rA * ND + kk + koff];
      a.y = emb[rA * ND + kk + koff + 1];
#pragma unroll
      for (int q = 0; q < 4; ++q) {
        v2f b;
        b.x = emb[nB[q] * ND + kk + koff];
        b.y = emb[nB[q] * ND + kk + koff + 1];
        acc[q] = __builtin_amdgcn_wmma_f32_16x16x4_f32(false, a, false, b, (short)0, acc[q], false, false);
      }
    }
#pragma unroll
    for (int q = 0; q < 4; ++q)
#pragma unroll
      for (int g = 0; g < 8; ++g)
        cosd[(rowbase + g) * NN + j0 + q * 16 + col0] = acc[q][g];
  }

  // ---- cov: centered data @ data^T, K = 8192 ----
  {
    v8f acc[4] = {};
#pragma unroll 1
    for (int kk = 0; kk < NC; kk += 4) {
      int c = kk + koff;
      int bb = c >> 6, w = c & 63;          // w even -> w, w+1 in same row
      const float* pA = data + (bb * NN + rA) * NW + w;
      v2f a;
      a.x = pA[0] - meanA; a.y = pA[1] - meanA;
#pragma unroll
      for (int q = 0; q < 4; ++q) {
        const float* pB = data + (bb * NN + nB[q]) * NW + w;
        v2f b;
        b.x = pB[0] - meanB[q]; b.y = pB[1] - meanB[q];
        acc[q] = __builtin_amdgcn_wmma_f32_16x16x4_f32(false, a, false, b, (short)0, acc[q], false, false);
      }
    }
#pragma unroll
    for (int q = 0; q < 4; ++q)
#pragma unroll
      for (int g = 0; g < 8; ++g)
        cov[(rowbase + g) * NN + j0 + q * 16 + col0] = acc[q][g];
  }
}

// ---------------------------------------------------------------------------
// 3) std from diag(cov)
__global__ void gdn_std(const float* __restrict__ cov, float* __restrict__ stdv) {
  int i = threadIdx.x;
  stdv[i] = sqrtf(cov[i * NN + i] + 1e-8f);
}

// ---------------------------------------------------------------------------
// 4) blend sim + top-20 per row (stable: ties -> smaller index)
__global__ void gdn_topk(const float* __restrict__ cosd, const float* __restrict__ cov,
                         const float* __restrict__ nrm, const float* __restrict__ stdv,
                         int* __restrict__ topk) {
  __shared__ float v[256];
  __shared__ float rv[256];
  __shared__ int   ri[256];
  const int i = blockIdx.x, t = threadIdx.x;
  float cosv = cosd[i * NN + t] / (nrm[i] * nrm[t] + 1e-8f);
  float corr = cov [i * NN + t] / (stdv[i] * stdv[t] + 1e-8f);
  v[t] = 0.7f * cosv + 0.3f * corr;
  __syncthreads();
  for (int k = 0; k < NK; ++k) {
    rv[t] = v[t]; ri[t] = t;
    __syncthreads();
    for (int off = 128; off > 0; off >>= 1) {
      if (t < off) {
        float vo = rv[t + off]; int io = ri[t + off];
        if (vo > rv[t] || (vo == rv[t] && io < ri[t])) { rv[t] = vo; ri[t] = io; }
      }
      __syncthreads();
    }
    if (t == 0) { topk[i * NK + k] = ri[0]; v[ri[0]] = -INFINITY; }
    __syncthreads();
  }
}

// ---------------------------------------------------------------------------
// 5) xl = x @ lin_W : [32768 x 64] @ [64 x 128].
//    lin_W staged in LDS (32 KB); each wave computes a 16-row x 128-col strip
//    (8 accumulators), one global A fragment per K-step reused across 8 WMMAs.
__global__ void gdn_lin(const float* __restrict__ data, const float* __restrict__ linW,
                        float* __restrict__ xl) {
  __shared__ float sW[NW * ND];                       // 64x128 = 32 KB
  const int t = threadIdx.x;
#pragma unroll
  for (int c = 0; c < NW * ND / 256; ++c)
    sW[c * 256 + t] = linW[c * 256 + t];
  __syncthreads();

  const int wave = t >> 5;                            // 0..7
  const int lane = t & 31;
  const int r0 = (blockIdx.x * 8 + wave) * 16;        // 2048 row tiles total
  const int m = lane & 15;
  const int koff = (lane >> 4) << 1;
  const float* xrow = data + (r0 + m) * NW;           // x rows contiguous [B*N, W]

  v8f acc[8] = {};
#pragma unroll 1
  for (int kk = 0; kk < NW; kk += 4) {
    v2f a;
    a.x = xrow[kk + koff];
    a.y = xrow[kk + koff + 1];
#pragma unroll
    for (int q = 0; q < 8; ++q) {
      v2f b;
      b.x = sW[(kk + koff) * ND + q * 16 + m];
      b.y = sW[(kk + koff + 1) * ND + q * 16 + m];
      acc[q] = __builtin_amdgcn_wmma_f32_16x16x4_f32(false, a, false, b, (short)0, acc[q], false, false);
    }
  }
  const int rowbase = r0 + ((lane >> 4) << 3);
  const int col0 = lane & 15;
#pragma unroll
  for (int q = 0; q < 8; ++q)
#pragma unroll
    for (int g = 0; g < 8; ++g)
      xl[(rowbase + g) * ND + q * 16 + col0] = acc[q][g];
}

// ---------------------------------------------------------------------------
// 6) per-node attention scores s_i, s_j (one wave per node, 4 dims per lane)
__global__ void gdn_scores(const float* __restrict__ xl, const float* __restrict__ emb,
                           const float* __restrict__ atti, const float* __restrict__ attj,
                           const float* __restrict__ attemi, const float* __restrict__ attemj,
                           float* __restrict__ s_i, float* __restrict__ s_j) {
  const int p = (blockIdx.x * blockDim.x + threadIdx.x) >> 5;  // 0..32767
  const int lane = threadIdx.x & 31;
  const int n = p & (NN - 1);
  float si = 0.f, sj = 0.f;
  const int d0 = lane * 4;
#pragma unroll
  for (int q = 0; q < 4; ++q) {
    int d = d0 + q;
    float xv = xl[p * ND + d];
    float ev = emb[n * ND + d];
    si += xv * atti[d] + ev * attemi[d];
    sj += xv * attj[d] + ev * attemj[d];
  }
  for (int off = 16; off > 0; off >>= 1) {
    si += __shfl_xor(si, off, 32);
    sj += __shfl_xor(sj, off, 32);
  }
  if (lane == 0) { s_i[p] = si; s_j[p] = sj; }
}

// ---------------------------------------------------------------------------
// 7) fixed fan-in (K=20) softmax aggregation + BN1/ReLU + emb scale + BN2/ReLU
//    + output projection. One wave per (b, i).
__global__ void gdn_attn(const float* __restrict__ xl, const float* __restrict__ emb,
                         const int* __restrict__ topk,
                         const float* __restrict__ s_i, const float* __restrict__ s_j,
                         const float* __restrict__ gbias,
                         const float* __restrict__ g1, const float* __restrict__ b1,
                         const float* __restrict__ g2, const float* __restrict__ b2,
                         const float* __restrict__ outW, const float* __restrict__ outb,
                         float* __restrict__ y) {
  const int seg = (blockIdx.x * blockDim.x + threadIdx.x) >> 5;  // b*N + i
  const int lane = threadIdx.x & 31;
  const int b = seg >> 8;
  const int i = seg & (NN - 1);

  int j = 0;
  float alpha = -INFINITY;
  if (lane < NK) {
    j = topk[i * NK + lane];
    float a = s_i[seg] + s_j[b * NN + j];
    alpha = (a > 0.f) ? a : 0.2f * a;           // leaky_relu(0.2)
  }
  float amax = alpha;
  for (int off = 16; off > 0; off >>= 1) amax = fmaxf(amax, __shfl_xor(amax, off, 32));
  float ex = (lane < NK) ? __expf(alpha - amax) : 0.f;
  float den = ex;
  for (int off = 16; off > 0; off >>= 1) den += __shfl_xor(den, off, 32);
  float w = ex / den;

  float4 acc = {0.f, 0.f, 0.f, 0.f};
#pragma unroll 1
  for (int k = 0; k < NK; ++k) {
    float wk = __shfl(w, k, 32);
    int jk = __shfl(j, k, 32);
    const float4 xv = *reinterpret_cast<const float4*>(xl + (b * NN + jk) * ND + lane * 4);
    acc.x += wk * xv.x; acc.y += wk * xv.y; acc.z += wk * xv.z; acc.w += wk * xv.w;
  }

  const float inv = rsqrtf(1.0f + 1e-5f);       // eval BN with running stats (0,1)
  const int d0 = lane * 4;
  float o[4] = {acc.x, acc.y, acc.z, acc.w};
  float part = 0.f;
#pragma unroll
  for (int q = 0; q < 4; ++q) {
    int d = d0 + q;
    float o1 = o[q] + gbias[d];
    o1 = o1 * inv * g1[d] + b1[d];
    o1 = (o1 > 0.f) ? o1 : 0.f;                 // ReLU
    float h = o1 * emb[i * ND + d];
    h = h * inv * g2[d] + b2[d];
    h = (h > 0.f) ? h : 0.f;                    // ReLU
    part += h * outW[d];
  }
  for (int off = 16; off > 0; off >>= 1) part += __shfl_xor(part, off, 32);
  if (lane == 0) y[seg] = part + outb[0];
}

// ---------------------------------------------------------------------------
extern "C" void kernel_launch(void* const* d_in, const int* in_sizes, int n_in,
                              void* d_out, int out_size, void* d_ws, size_t ws_size,
                              hipStream_t stream) {
  const float* data   = (const float*)d_in[0];
  // d_in[1] = org_edge_index (unused by forward)
  const float* emb    = (const float*)d_in[2];
  const float* linW   = (const float*)d_in[3];
  const float* atti   = (const float*)d_in[4];
  const float* attj   = (const float*)d_in[5];
  const float* attemi = (const float*)d_in[6];
  const float* attemj = (const float*)d_in[7];
  const float* gbias  = (const float*)d_in[8];
  const float* g1     = (const float*)d_in[9];
  const float* b1     = (const float*)d_in[10];
  const float* g2     = (const float*)d_in[11];
  const float* b2     = (const float*)d_in[12];
  const float* outW   = (const float*)d_in[13];
  const float* outb   = (const float*)d_in[14];
  float* y = (float*)d_out;

  // workspace layout (floats); xl offset is 16B-aligned for float4 loads
  float* ws   = (float*)d_ws;
  float* nrm  = ws;                    // 256
  float* mean = nrm + NN;              // 256
  float* stdv = mean + NN;             // 256
  float* cosd = stdv + NN;             // 65536
  float* cov  = cosd + NN * NN;        // 65536
  int*   topk = (int*)(cov + NN * NN); // 5120
  float* xl   = (float*)(topk + NN * NK);  // 32768*128
  float* s_i  = xl + NB * NN * ND;     // 32768
  float* s_j  = s_i + NB * NN;         // 32768

  gdn_stats  <<<NN,   256, 0, stream>>>(data, emb, mean, nrm);
  gdn_simgemm<<<64,    32, 0, stream>>>(data, emb, mean, cosd, cov);
  gdn_std    <<<1,    256, 0, stream>>>(cov, stdv);
  gdn_topk   <<<NN,   256, 0, stream>>>(cosd, cov, nrm, stdv, topk);
  gdn_lin    <<<256,  256, 0, stream>>>(data, linW, xl);
  gdn_scores <<<4096, 256, 0, stream>>>(xl, emb, atti, attj, attemi, attemj, s_i, s_j);
  gdn_attn   <<<4096, 256, 0, stream>>>(xl, emb, topk, s_i, s_j, gbias,
                                        g1, b1, g2, b2, outW, outb, y);
}